// RNN_10952166605437
// MI455X (gfx1250) — compile-verified
//
#include <hip/hip_runtime.h>
#include <math.h>

// ---------------- problem constants ----------------
#define DIM   512          // H == S == U == Z
#define BB    64
#define TT    1024
#define ALPHA 0.1f
#define LDSW  516          // 512 + 4 pad: keeps 16B alignment (516*4 % 16 == 0), spreads banks

typedef __attribute__((ext_vector_type(2))) float v2f;
typedef __attribute__((ext_vector_type(8))) float v8f;

// ---- CDNA5 async copy: global -> LDS, 16B per lane, tracked by ASYNCcnt ----
__device__ __forceinline__ void async_ld_b128(unsigned lds_off, const float* g) {
  asm volatile("global_load_async_to_lds_b128 %0, %1, off"
               :: "v"(lds_off), "v"(g) : "memory");
}
__device__ __forceinline__ void wait_async0() {
  asm volatile("s_wait_asynccnt 0" ::: "memory");
}

// Stage 16 rows [n0, n0+16) of row-major Wt[N][512] into lds[16][LDSW] (float4 copies).
__device__ __forceinline__ void stage_rows16(const float* __restrict__ Wt, int n0,
                                             float* lds, int tid, int nthreads) {
  for (int i = tid; i < 16 * (DIM / 4); i += nthreads) {
    int row = i >> 7;           // / 128 float4 per row
    int c   = (i & 127) << 2;   // * 4
    float4 v = *(const float4*)(Wt + (size_t)(n0 + row) * DIM + c);
    *(float4*)(lds + row * LDSW + c) = v;
  }
}

// One 16x16 fp32 tile of  A[m0:m0+16, 0:512] @ Wt[n0:n0+16, 0:512]^T  via V_WMMA_F32_16X16X4_F32.
// A is row-major with row stride lda (global or LDS; addrspace inferred after inlining);
// Wt rows pre-staged in lds[16][LDSW].
__device__ __forceinline__ v8f wmma_gemm_512(const float* __restrict__ A, size_t lda,
                                             int m0, const float* lds, v8f acc, int lane) {
  const int half = lane >> 4;
  const int lr   = lane & 15;
  const float* arow = A + (size_t)(m0 + lr) * lda + 2 * half;
  const float* brow = lds + lr * LDSW + 2 * half;
  #pragma unroll 8
  for (int kb = 0; kb < DIM; kb += 4) {
    v2f a = *(const v2f*)(arow + kb);
    v2f b = *(const v2f*)(brow + kb);
    acc = __builtin_amdgcn_wmma_f32_16x16x4_f32(false, a, false, b, (short)0, acc,
                                                false, false);
  }
  return acc;
}

// ---------------- kernels ----------------

// Zero the grid-barrier counter and copy h0 into h_seq[:, 0, :].
__global__ void rnn_init_kernel(const float* __restrict__ h0, float* __restrict__ hseq,
                                unsigned* __restrict__ cnt) {
  int idx = blockIdx.x * blockDim.x + threadIdx.x;
  if (idx == 0) *cnt = 0u;
  for (int i = idx; i < BB * DIM; i += gridDim.x * blockDim.x) {
    int b = i >> 9;
    int h = i & (DIM - 1);
    hseq[(size_t)b * TT * DIM + h] = h0[i];
  }
}

// drive[m, n] = sum_k s[m,k]*Bs[n,k] + sum_k u[m,k]*Bu[n,k] + bias[n],  m = b*T + t
__global__ void rnn_drive_kernel(const float* __restrict__ s_seq, const float* __restrict__ u_seq,
                                 const float* __restrict__ Bsm, const float* __restrict__ Bum,
                                 const float* __restrict__ bias, float* __restrict__ drive) {
  __shared__ float ldsB[16 * LDSW];
  const int tid  = threadIdx.x;
  const int lane = tid & 31;
  const int wave = tid >> 5;                       // 0..7
  const int n0   = blockIdx.y * 16;
  const int m0   = (blockIdx.x * 8 + wave) * 16;

  v8f acc = {};
  stage_rows16(Bsm, n0, ldsB, tid, blockDim.x);
  __syncthreads();
  acc = wmma_gemm_512(s_seq, DIM, m0, ldsB, acc, lane);
  __syncthreads();
  stage_rows16(Bum, n0, ldsB, tid, blockDim.x);
  __syncthreads();
  acc = wmma_gemm_512(u_seq, DIM, m0, ldsB, acc, lane);

  const int half = lane >> 4, lr = lane & 15;
  const float bv = bias[n0 + lr];
  #pragma unroll
  for (int r = 0; r < 8; ++r) {
    int m = m0 + r + 8 * half;                     // D layout: lane<16 -> M=r, lane>=16 -> M=8+r
    drive[(size_t)m * DIM + n0 + lr] = acc[r] + bv;
  }
}

// Persistent sequential scan: 32 blocks (one 16-col tile each) x 4 waves (16-row batch tile each).
// Per step: async-stage h_t[64][512] (128KB) into LDS (coalesced, ASYNCcnt-tracked),
// WMMA from LDS, fused leaky-tanh epilogue, device-scope grid barrier.
__global__ void __launch_bounds__(128, 1)
rnn_scan_kernel(float* __restrict__ hseq, const float* __restrict__ drive,
                const float* __restrict__ W, unsigned* __restrict__ cnt) {
  __shared__ float Wlds[16 * LDSW];                // 33 KB: W rows j0..j0+15
  __shared__ float Hlds[BB * LDSW];                // 132 KB: full h_t, all 64 batches
  const int tid  = threadIdx.x;
  const int lane = tid & 31;
  const int wave = tid >> 5;                       // 0..3
  const int j0   = blockIdx.x * 16;                // output column tile
  const int b0   = wave * 16;                      // batch row tile
  const int half = lane >> 4, lr = lane & 15;

  stage_rows16(W, j0, Wlds, tid, blockDim.x);      // resident for all 1023 steps
  const unsigned hbase = (unsigned)(size_t)&Hlds[0];  // LDS byte offset of Hlds
  __syncthreads();

  for (int t = 0; t < TT - 1; ++t) {
    // ---- async-stage h_t: row b lives at hseq[(b*T + t)*DIM] ----
    {
      const float* src = hseq + (size_t)t * DIM;
      for (int i = tid; i < BB * (DIM / 4); i += blockDim.x) {
        int row = i >> 7;                          // batch index (one full row per iteration set)
        int c   = (i & 127) << 2;                  // coalesced float4 within the row
        async_ld_b128(hbase + (unsigned)(row * LDSW + c) * 4u,
                      src + (size_t)row * TT * DIM + c);
      }
      wait_async0();
    }
    __syncthreads();

    v8f acc = {};
    acc = wmma_gemm_512(Hlds, LDSW, b0, Wlds, acc, lane);   // A-frags + B-frags from LDS

    #pragma unroll
    for (int r = 0; r < 8; ++r) {
      int b = b0 + r + 8 * half;
      float pre = Hlds[b * LDSW + j0 + lr];        // h_t from LDS
      float dv  = drive[((size_t)b * TT + t) * DIM + j0 + lr];
      float hn  = (1.0f - ALPHA) * pre + ALPHA * tanhf(acc[r] + dv);
      hseq[((size_t)b * TT + t + 1) * DIM + j0 + lr] = hn;  // write h at t+1
    }

    // ---- device-scope grid barrier (all 32 blocks resident) ----
    __threadfence();                               // make this thread's h_{t+1} stores visible
    __syncthreads();
    if (tid == 0) {
      __hip_atomic_fetch_add(cnt, 1u, __ATOMIC_RELEASE, __HIP_MEMORY_SCOPE_AGENT);
      const unsigned target = (unsigned)gridDim.x * (unsigned)(t + 1);
      while (__hip_atomic_load(cnt, __ATOMIC_ACQUIRE, __HIP_MEMORY_SCOPE_AGENT) < target)
        __builtin_amdgcn_s_sleep(1);
    }
    __syncthreads();
    __threadfence();                               // acquire: invalidate stale cached h data
  }
}

// z[m, n] = sum_k hseq[m,k] * Wz[n,k] + bz[n]   (overwrites the drive scratch region)
__global__ void rnn_z_kernel(const float* __restrict__ hseq, const float* __restrict__ Wz,
                             const float* __restrict__ bz, float* __restrict__ z) {
  __shared__ float ldsW[16 * LDSW];
  const int tid  = threadIdx.x;
  const int lane = tid & 31;
  const int wave = tid >> 5;
  const int n0   = blockIdx.y * 16;
  const int m0   = (blockIdx.x * 8 + wave) * 16;

  stage_rows16(Wz, n0, ldsW, tid, blockDim.x);
  __syncthreads();
  v8f acc = {};
  acc = wmma_gemm_512(hseq, DIM, m0, ldsW, acc, lane);

  const int half = lane >> 4, lr = lane & 15;
  const float bv = bz[n0 + lr];
  #pragma unroll
  for (int r = 0; r < 8; ++r) {
    int m = m0 + r + 8 * half;
    z[(size_t)m * DIM + n0 + lr] = acc[r] + bv;
  }
}

// ---------------- host launcher ----------------
extern "C" void kernel_launch(void* const* d_in, const int* in_sizes, int n_in,
                              void* d_out, int out_size, void* d_ws, size_t ws_size,
                              hipStream_t stream) {
  const float* h0    = (const float*)d_in[0];
  const float* s_seq = (const float*)d_in[1];
  const float* u_seq = (const float*)d_in[2];
  const float* W     = (const float*)d_in[3];
  const float* bvec  = (const float*)d_in[4];
  const float* Bsm   = (const float*)d_in[5];
  const float* Bum   = (const float*)d_in[6];
  const float* Wz    = (const float*)d_in[7];
  const float* bz    = (const float*)d_in[8];

  float* hseq = (float*)d_out;                          // [B, T, H]
  float* zreg = hseq + (size_t)BB * TT * DIM;           // [B, T, Z]: used as drive scratch, then z
  unsigned* cnt = (unsigned*)d_ws;                      // grid-barrier counter (re-zeroed each launch)

  rnn_init_kernel<<<64, 256, 0, stream>>>(h0, hseq, cnt);
  rnn_drive_kernel<<<dim3(512, 32), 256, 0, stream>>>(s_seq, u_seq, Bsm, Bum, bvec, zreg);
  rnn_scan_kernel<<<32, 128, 0, stream>>>(hseq, zreg, W, cnt);
  rnn_z_kernel<<<dim3(512, 32), 256, 0, stream>>>(hseq, Wz, bz, zreg);
}